// FVNet_27419071217689
// MI455X (gfx1250) — compile-verified
//
#include <hip/hip_runtime.h>
#include <math.h>

// Problem constants (from reference): BATCH, FEAT, NUM, K = (8, 512, 512, 32)
#define BATCH 8
#define FEAT  512
#define NUMD  512
#define KD    32

typedef __attribute__((ext_vector_type(2))) float v2f;
typedef __attribute__((ext_vector_type(8))) float v8f;

// CDNA5 f32 WMMA: D(16x16,f32) = A(16x4,f32) * B(4x16,f32) + C
// 8-arg form: (neg_a, A, neg_b, B, c_mod, C, reuse_a, reuse_b)
static __device__ __forceinline__ v8f wmma_f32_16x16x4(v2f a, v2f b, v8f c) {
  return __builtin_amdgcn_wmma_f32_16x16x4_f32(false, a, false, b, (short)0, c,
                                               false, false);
}

// ---------------------------------------------------------------------------
// Kernel 0: zero the atomic accumulators (asum[B*K] followed by norms[2*B]).
// ---------------------------------------------------------------------------
__global__ void fv_init(float* __restrict__ acc) {
  const int t = threadIdx.x;
  if (t < BATCH * KD + 2 * BATCH) acc[t] = 0.0f;
}

// ---------------------------------------------------------------------------
// Kernel 1: logits = xt @ W^T + b  (WMMA f32 16x16x4), softmax over K, asum.
// Grid: (N/128, B), block 256 (8 waves). Each wave owns one 16-row n-tile and
// computes BOTH 16-wide k-tiles, sharing a single A fragment per step.
// A-frag layout (ISA 7.12.2, 32-bit A 16x4): lane<16 -> {K=0,K=1},
// lane>=16 -> {K=2,K=3}; M = lane&15.  B-frag symmetric over rows of B.
// ---------------------------------------------------------------------------
__global__ void __launch_bounds__(256)
fv_logits_softmax(const float* __restrict__ x, const float* __restrict__ W,
                  const float* __restrict__ bias, float* __restrict__ a_out,
                  float* __restrict__ asum) {
  __shared__ float sm[128 * KD];  // logits then softmaxed 'a' for 128 rows
  const int b    = blockIdx.y;
  const int n0   = blockIdx.x * 128;
  const int tid  = threadIdx.x;
  const int wave = tid >> 5;
  const int lane = tid & 31;
  const int half = lane >> 4;   // which K-pair of the 16x4 A fragment
  const int lm   = lane & 15;
  const int row_n = n0 + wave * 16 + lm;  // M index (row of xt = column of x)

  const float* xb  = x + (size_t)b * FEAT * NUMD;
  const float* wr0 = W + (size_t)lm * FEAT;         // W row for k = lm
  const float* wr1 = W + (size_t)(16 + lm) * FEAT;  // W row for k = 16+lm

  v8f acc0 = {}, acc1 = {};
#pragma unroll 4
  for (int f = 0; f < FEAT; f += 4) {
    const int fc = f + 2 * half;
    v2f aF, bF0, bF1;
    aF.x = xb[(size_t)(fc + 0) * NUMD + row_n];  // xt[n][f] = x[f][n]
    aF.y = xb[(size_t)(fc + 1) * NUMD + row_n];
    const float2 wp0 = *(const float2*)(wr0 + fc);
    const float2 wp1 = *(const float2*)(wr1 + fc);
    bF0.x = wp0.x; bF0.y = wp0.y;
    bF1.x = wp1.x; bF1.y = wp1.y;
    acc0 = wmma_f32_16x16x4(aF, bF0, acc0);   // k = 0..15
    acc1 = wmma_f32_16x16x4(aF, bF1, acc1);   // k = 16..31
  }

  // D layout: lane<16 -> rows M=r (vgpr r), lane>=16 -> rows M=8+r.
  const float bv0 = bias[lm];
  const float bv1 = bias[16 + lm];
#pragma unroll
  for (int r = 0; r < 8; ++r) {
    const int row = wave * 16 + r + 8 * half;
    sm[row * KD + lm]      = acc0[r] + bv0;
    sm[row * KD + 16 + lm] = acc1[r] + bv1;
  }
  __syncthreads();

  // Softmax over K=32 per row; 128 rows handled by 128 threads.
  if (tid < 128) {
    float v[KD];
    float mx = -INFINITY;
#pragma unroll
    for (int k = 0; k < KD; ++k) {
      v[k] = sm[tid * KD + k];
      mx = fmaxf(mx, v[k]);
    }
    float s = 0.0f;
#pragma unroll
    for (int k = 0; k < KD; ++k) {
      v[k] = __expf(v[k] - mx);
      s += v[k];
    }
    const float inv = 1.0f / s;
    float* ar = a_out + ((size_t)b * NUMD + n0 + tid) * KD;
#pragma unroll
    for (int k = 0; k < KD; ++k) {
      const float av = v[k] * inv;
      ar[k] = av;
      sm[tid * KD + k] = av;  // reuse LDS for the asum reduction
    }
  }
  __syncthreads();

  // Partial asum over this 128-row strip: one atomic per k per block.
  if (tid < KD) {
    float s = 0.0f;
    for (int r = 0; r < 128; ++r) s += sm[r * KD + tid];
    atomicAdd(&asum[b * KD + tid], s);
  }
}

// ---------------------------------------------------------------------------
// Kernel 2: ax = x @ a, ax2 = (x*x) @ a.  Four WMMA chains per wave (ax/ax2 x
// two k-tiles) all sharing one float2 A load per 4-deep step.
// Grid: (F/128, B), block 256 (8 waves). Wave w owns one 16-row f-tile.
// ---------------------------------------------------------------------------
__global__ void __launch_bounds__(256)
fv_weighted_moments(const float* __restrict__ x, const float* __restrict__ a_mat,
                    float* __restrict__ ax, float* __restrict__ ax2) {
  const int b    = blockIdx.y;
  const int tid  = threadIdx.x;
  const int wave = tid >> 5;
  const int lane = tid & 31;
  const int half = lane >> 4;
  const int lm   = lane & 15;
  const int f0   = blockIdx.x * 128 + wave * 16;
  const int frow = f0 + lm;  // M index (feature row)

  const float* xr  = x + ((size_t)b * FEAT + frow) * NUMD;       // row of x[b]
  const float* am0 = a_mat + (size_t)b * NUMD * KD + lm;         // col k=lm
  const float* am1 = am0 + 16;                                   // col k=16+lm

  v8f accx0 = {}, accx1 = {}, accq0 = {}, accq1 = {};
#pragma unroll 4
  for (int n = 0; n < NUMD; n += 4) {
    const int nc = n + 2 * half;
    const float2 xp = *(const float2*)(xr + nc);  // x[f][n], x[f][n+1]
    v2f aF, a2F, bF0, bF1;
    aF.x = xp.x;
    aF.y = xp.y;
    a2F.x = xp.x * xp.x;
    a2F.y = xp.y * xp.y;
    bF0.x = am0[(size_t)(nc + 0) * KD];  // a[n][lm]
    bF0.y = am0[(size_t)(nc + 1) * KD];
    bF1.x = am1[(size_t)(nc + 0) * KD];  // a[n][16+lm]
    bF1.y = am1[(size_t)(nc + 1) * KD];
    accx0 = wmma_f32_16x16x4(aF,  bF0, accx0);
    accx1 = wmma_f32_16x16x4(aF,  bF1, accx1);
    accq0 = wmma_f32_16x16x4(a2F, bF0, accq0);
    accq1 = wmma_f32_16x16x4(a2F, bF1, accq1);
  }

#pragma unroll
  for (int r = 0; r < 8; ++r) {
    const int f = f0 + r + 8 * half;
    const size_t base = ((size_t)b * FEAT + f) * KD;
    ax [base + lm]      = accx0[r];
    ax [base + 16 + lm] = accx1[r];
    ax2[base + lm]      = accq0[r];
    ax2[base + 16 + lm] = accq1[r];
  }
}

// ---------------------------------------------------------------------------
// Kernel 3: fv1/fv2 per (b,f) row, row L2 norm, write out, accumulate global
// squared norms per (b, fv-half) with one atomic per wave.
// Grid: (F/256, B), block 256.
// ---------------------------------------------------------------------------
__global__ void __launch_bounds__(256)
fv_finalize_rows(const float* __restrict__ ax, const float* __restrict__ ax2m,
                 const float* __restrict__ asum, const float* __restrict__ mu,
                 const float* __restrict__ sigma, float* __restrict__ out,
                 float* __restrict__ norms) {
  __shared__ float s_asum[KD];
  const int b   = blockIdx.y;
  const int tid = threadIdx.x;
  if (tid < KD) s_asum[tid] = asum[b * KD + tid];
  __syncthreads();

  const int f = blockIdx.x * 256 + tid;
  const float* axr  = ax   + ((size_t)b * FEAT + f) * KD;
  const float* ax2r = ax2m + ((size_t)b * FEAT + f) * KD;
  const float* mur  = mu    + (size_t)f * KD;
  const float* sgr  = sigma + (size_t)f * KD;

  float v1a[KD], v2a[KD];
  float ss1 = 0.0f, ss2 = 0.0f;
#pragma unroll
  for (int k = 0; k < KD; ++k) {
    const float axv = axr[k], ax2v = ax2r[k];
    const float m = mur[k], sg = sgr[k], as = s_asum[k];
    const float v1 = (axv - m * as) / sg;
    const float v2 = (ax2v - 2.0f * m * axv + m * m * as) / (sg * sg) - as;
    v1a[k] = v1;
    v2a[k] = v2;
    ss1 += v1 * v1;
    ss2 += v2 * v2;
  }
  const float i1 = 1.0f / fmaxf(sqrtf(ss1), 1e-12f);
  const float i2 = 1.0f / fmaxf(sqrtf(ss2), 1e-12f);

  float g1 = 0.0f, g2 = 0.0f;
  float* o1 = out + (size_t)b * (2 * FEAT * KD) + (size_t)f * KD;
  float* o2 = o1 + FEAT * KD;
#pragma unroll
  for (int k = 0; k < KD; ++k) {
    const float w1 = v1a[k] * i1;
    const float w2 = v2a[k] * i2;
    o1[k] = w1;
    o2[k] = w2;
    g1 += w1 * w1;
    g2 += w2 * w2;
  }

  // wave32 shuffle reduction, then one atomic per wave per accumulator
#pragma unroll
  for (int off = 16; off > 0; off >>= 1) {
    g1 += __shfl_down(g1, off);
    g2 += __shfl_down(g2, off);
  }
  if ((tid & 31) == 0) {
    atomicAdd(&norms[2 * b + 0], g1);
    atomicAdd(&norms[2 * b + 1], g2);
  }
}

// ---------------------------------------------------------------------------
// Kernel 4: global L2 scale of the concatenated output.
// out is (B, 2*F*K): first F*K = fv1, second F*K = fv2, each per-batch.
// ---------------------------------------------------------------------------
__global__ void __launch_bounds__(256)
fv_global_scale(float* __restrict__ out, const float* __restrict__ norms) {
  const int idx = blockIdx.x * 256 + threadIdx.x;  // 0 .. B*2*F*K-1
  const int b = idx >> 15;          // 2*F*K = 32768 per batch
  const int h = (idx >> 14) & 1;    // F*K = 16384 per fv half
  const float nr = norms[b * 2 + h];
  out[idx] *= 1.0f / fmaxf(sqrtf(nr), 1e-12f);
}

// ---------------------------------------------------------------------------
// Host-side launcher
// ---------------------------------------------------------------------------
extern "C" void kernel_launch(void* const* d_in, const int* in_sizes, int n_in,
                              void* d_out, int out_size, void* d_ws, size_t ws_size,
                              hipStream_t stream) {
  const float* x     = (const float*)d_in[0];  // (B, F, N)
  const float* W     = (const float*)d_in[1];  // (K, F)
  const float* bias  = (const float*)d_in[2];  // (K,)
  const float* mu    = (const float*)d_in[3];  // (F, K)
  const float* sigma = (const float*)d_in[4];  // (F, K)
  float* out = (float*)d_out;                  // (B, 2*F*K)
  float* ws  = (float*)d_ws;

  // Workspace layout (floats):
  float* a_mat = ws;                    // B*N*K = 131072
  float* ax    = ws + 131072;           // B*F*K = 131072
  float* ax2   = ws + 262144;           // B*F*K = 131072
  float* asum  = ws + 393216;           // B*K   = 256
  // norms = asum + 256 (2*B = 16 floats), zeroed together with asum
  float* norms = ws + 393472;

  fv_init<<<1, 512, 0, stream>>>(asum);
  fv_logits_softmax<<<dim3(NUMD / 128, BATCH), 256, 0, stream>>>(x, W, bias,
                                                                 a_mat, asum);
  fv_weighted_moments<<<dim3(FEAT / 128, BATCH), 256, 0, stream>>>(x, a_mat,
                                                                   ax, ax2);
  fv_finalize_rows<<<dim3(FEAT / 256, BATCH), 256, 0, stream>>>(ax, ax2, asum,
                                                                mu, sigma, out,
                                                                norms);
  fv_global_scale<<<(BATCH * 2 * FEAT * KD) / 256, 256, 0, stream>>>(out, norms);
}